// LearnedAttention_68865505624686
// MI455X (gfx1250) — compile-verified
//
#include <hip/hip_runtime.h>
#include <hip/hip_bf16.h>

typedef __attribute__((ext_vector_type(2))) float v2f;
typedef __attribute__((ext_vector_type(8))) float v8f;

#define NB     32      // batch
#define SS     4096    // sequence (keys)
#define HH     8       // heads
#define DD     64      // head dim
#define HD     512     // H*D flattened value row
#define NCHUNK 8       // s-chunks for AV kernel
#define SCHUNK 512     // S / NCHUNK
#define APITCH 513     // LDS pitch for 8x512 prob tile (kills bank conflicts)

// ---------------------------------------------------------------------------
// Kernel 1: k_score[n,h,s] = dot(keys[n,s,h,:], wk).  One wave per (n,s) row.
// Lanes hold float2 of each head's 64 floats -> coalesced 256B reads.
// ---------------------------------------------------------------------------
__global__ void __launch_bounds__(256)
score_kernel(const float* __restrict__ K, const float* __restrict__ wk,
             float* __restrict__ scores) {
    const int tid  = threadIdx.x;
    const int lane = tid & 31;
    const int w    = tid >> 5;
    const long row = (long)blockIdx.x * 8 + w;       // n*S + s
    const int  n   = (int)(row >> 12);               // / 4096
    const int  s   = (int)(row & 4095);
    const float* kr = K + (size_t)row * HD;

    const float wkx = wk[2 * lane];
    const float wky = wk[2 * lane + 1];

#pragma unroll
    for (int h = 0; h < HH; ++h) {
        float2 kv = *(const float2*)(kr + h * DD + lane * 2);
        float p = kv.x * wkx + kv.y * wky;
#pragma unroll
        for (int off = 16; off > 0; off >>= 1)
            p += __shfl_xor(p, off, 32);
        if (lane == 0)
            scores[((size_t)(n * HH + h)) * SS + s] = p;
    }
}

// ---------------------------------------------------------------------------
// Kernel 2: per-(n,h) softmax stats: max and 1/sum(exp).
// ---------------------------------------------------------------------------
__global__ void __launch_bounds__(256)
stats_kernel(const float* __restrict__ scores, float* __restrict__ stats) {
    const int row = blockIdx.x;                       // n*8 + h
    const int tid = threadIdx.x;
    const float* sr = scores + (size_t)row * SS;
    __shared__ float red[256];

    float m = -3.4e38f;
    for (int s = tid; s < SS; s += 256) m = fmaxf(m, sr[s]);
    red[tid] = m; __syncthreads();
    for (int o = 128; o > 0; o >>= 1) {
        if (tid < o) red[tid] = fmaxf(red[tid], red[tid + o]);
        __syncthreads();
    }
    m = red[0]; __syncthreads();

    float sum = 0.f;
    for (int s = tid; s < SS; s += 256) sum += __expf(sr[s] - m);
    red[tid] = sum; __syncthreads();
    for (int o = 128; o > 0; o >>= 1) {
        if (tid < o) red[tid] += red[tid + o];
        __syncthreads();
    }
    if (tid == 0) { stats[2 * row] = m; stats[2 * row + 1] = 1.0f / red[0]; }
}

// ---------------------------------------------------------------------------
// Kernel 3: per (s-chunk, n): out_partial[16pad x 512] += A[16x512] @ V[512x512]
// using V_WMMA_F32_16X16X4_F32.  8 waves/block, wave owns 4 N-tiles of 16.
// A-frag (16x4 f32): lanes 0-15 -> M=lane, K={0,1}; lanes 16-31 -> M=lane-16,
// K={2,3}.  B-frag (4x16): VGPR0 = rows K={0|2}, VGPR1 = rows K={1|3}.
// ---------------------------------------------------------------------------
__global__ void __launch_bounds__(256)
av_wmma_kernel(const float* __restrict__ scores, const float* __restrict__ stats,
               const float* __restrict__ V, float* __restrict__ partial) {
    const int chunk = blockIdx.x;
    const int n     = blockIdx.y;
    const int tid   = threadIdx.x;
    const int s0    = chunk * SCHUNK;

    __shared__ float A_lds[HH * APITCH];

    // Stage probabilities for this s-chunk: A[l][k] = exp(score - m) * inv_sum
    for (int i = tid; i < HH * SCHUNK; i += 256) {
        const int l = i >> 9;            // / 512
        const int k = i & 511;
        const float m   = stats[2 * (n * HH + l)];
        const float inv = stats[2 * (n * HH + l) + 1];
        const float sc  = scores[((size_t)(n * HH + l)) * SS + s0 + k];
        A_lds[l * APITCH + k] = __expf(sc - m) * inv;
    }
    __syncthreads();

    const int lane = tid & 31;
    const int w    = tid >> 5;
    const int m15  = lane & 15;
    const int kh   = (lane < 16) ? 0 : 2;

    v8f acc[4];
#pragma unroll
    for (int t = 0; t < 4; ++t)
#pragma unroll
        for (int e = 0; e < 8; ++e) acc[t][e] = 0.f;

    const float* vb = V + ((size_t)n * SS + s0) * HD;

    for (int k0 = 0; k0 < SCHUNK; k0 += 4) {
        float ax = 0.f, ay = 0.f;
        if (m15 < 8) {                     // rows 8..15 are zero padding
            ax = A_lds[m15 * APITCH + k0 + kh];
            ay = A_lds[m15 * APITCH + k0 + kh + 1];
        }
        v2f a; a.x = ax; a.y = ay;
        const float* vrow = vb + (size_t)(k0 + kh) * HD;
#pragma unroll
        for (int t = 0; t < 4; ++t) {
            const int col = (w * 4 + t) * 16 + m15;
            v2f b; b.x = vrow[col]; b.y = vrow[HD + col];
            acc[t] = __builtin_amdgcn_wmma_f32_16x16x4_f32(
                false, a, false, b, (short)0, acc[t], false, false);
        }
    }

    // C/D layout: VGPR r -> M=r (lanes 0-15), M=r+8 (lanes 16-31).
    // Only M=0..7 are real rows -> lanes 0-15 write them out.
    if (lane < 16) {
#pragma unroll
        for (int t = 0; t < 4; ++t) {
            const int col = (w * 4 + t) * 16 + lane;
            float* pb = partial + (((size_t)chunk * NB + n) * HH) * HD + col;
#pragma unroll
            for (int l = 0; l < 8; ++l) pb[(size_t)l * HD] = acc[t][l];
        }
    }
}

// ---------------------------------------------------------------------------
// Kernel 4: deterministic fixed-order reduction of the 8 chunk partials.
// ---------------------------------------------------------------------------
__global__ void __launch_bounds__(256)
reduce_kernel(const float* __restrict__ partial, float* __restrict__ out) {
    const int i = blockIdx.x * 256 + threadIdx.x;
    float s = 0.f;
#pragma unroll
    for (int c = 0; c < NCHUNK; ++c)
        s += partial[(size_t)c * (NB * HH * HD) + i];
    out[i] = s;
}

// ---------------------------------------------------------------------------
extern "C" void kernel_launch(void* const* d_in, const int* in_sizes, int n_in,
                              void* d_out, int out_size, void* d_ws, size_t ws_size,
                              hipStream_t stream) {
    (void)in_sizes; (void)n_in; (void)out_size; (void)ws_size;
    // inputs: queries, keys, values, w, b  (queries & b cancel under softmax)
    const float* Kp = (const float*)d_in[1];
    const float* Vp = (const float*)d_in[2];
    const float* Wp = (const float*)d_in[3];
    float* out = (float*)d_out;

    float* ws      = (float*)d_ws;
    float* scores  = ws;                                   // 32*8*4096 = 1,048,576 f
    float* stats   = ws + (size_t)NB * HH * SS;            // 512 f (padded to 1024)
    float* partial = stats + 1024;                         // 8*32*8*512 = 1,048,576 f

    const float* wk = Wp + DD;                             // w[64:128]

    score_kernel <<<(NB * SS) / 8, 256, 0, stream>>>(Kp, wk, scores);
    stats_kernel <<<NB * HH,       256, 0, stream>>>(scores, stats);
    av_wmma_kernel<<<dim3(NCHUNK, NB), 256, 0, stream>>>(scores, stats, Vp, partial);
    reduce_kernel<<<(NB * HH * HD) / 256, 256, 0, stream>>>(partial, out);
}